// LPAnalysis_26147760898580
// MI455X (gfx1250) — compile-verified
//
#include <hip/hip_runtime.h>
#include <hip/hip_bf16.h>

// LPC residual (order-16 Levinson-Durbin + 17-tap FIR) for 128000 frames of 160
// f32 samples. Memory-bound (~164 MB traffic -> ~7us floor at 23.3 TB/s), so the
// kernel streams each frame through LDS exactly once. The autocorrelation is
// computed with the full-precision CDNA5 matrix op V_WMMA_F32_16X16X4_F32:
//   A[m][j] = f[4c+j+m], B[j][n] = f[4c+j]  =>  D[m][n] accumulates r[m].
// One wave32 per frame; 8 waves per block.

typedef float v2f __attribute__((ext_vector_type(2)));
typedef float v8f __attribute__((ext_vector_type(8)));

#define ORDER 16
#define FRAME 160
#define EPS_F 1e-8f
#define WAVES_PER_BLOCK 8
#define PAD 16
#define REGION (PAD + FRAME + PAD)  // 192 floats per wave

__global__ __launch_bounds__(256) void lpc_residual_kernel(
    const float* __restrict__ x, float* __restrict__ out, int n_frames) {
  __shared__ float sh[WAVES_PER_BLOCK][REGION];

  const int lane = threadIdx.x & 31;
  const int wv = threadIdx.x >> 5;
  const int frame = blockIdx.x * WAVES_PER_BLOCK + wv;
  const bool active = frame < n_frames;

  float* fr = &sh[wv][PAD];  // fr[-16..-1] and fr[160..175] are zero padding

  if (active) {
    if (lane < PAD) {
      sh[wv][lane] = 0.0f;                 // front pad (for FIR f[n-k])
      sh[wv][PAD + FRAME + lane] = 0.0f;   // end pad (for WMMA over-read)
    }
    const float* gx = x + (long long)frame * FRAME;
#pragma unroll
    for (int t = 0; t < 5; ++t) fr[lane + 32 * t] = gx[lane + 32 * t];
  }
  __syncthreads();  // all waves reach this; makes LDS frame visible cross-lane

  if (active) {
    // ---------------- Autocorrelation lags 0..15 via f32 WMMA ----------------
    // A (16x4 f32) lane layout: lane l -> row m=l&15, K = 2*(l>>4) + {v0,v1}.
    // B (4x16 f32): value depends only on K; K = 2*(l>>4) + {v0,v1} under both
    // plausible stripings, so this is layout-robust.
    const int m = lane & 15;
    const int k0 = (lane >> 4) << 1;  // 0 or 2
    v8f acc0 = {0.f, 0.f, 0.f, 0.f, 0.f, 0.f, 0.f, 0.f};
    v8f acc1 = acc0;
    for (int c = 0; c < 40; c += 2) {
      v2f A0, B0, A1, B1;
      A0.x = fr[4 * c + k0 + m];
      A0.y = fr[4 * c + k0 + 1 + m];
      B0.x = fr[4 * c + k0];
      B0.y = fr[4 * c + k0 + 1];
      A1.x = fr[4 * (c + 1) + k0 + m];
      A1.y = fr[4 * (c + 1) + k0 + 1 + m];
      B1.x = fr[4 * (c + 1) + k0];
      B1.y = fr[4 * (c + 1) + k0 + 1];
      // (neg_a, A, neg_b, B, c_mod, C, reuse_a, reuse_b)
      acc0 = __builtin_amdgcn_wmma_f32_16x16x4_f32(false, A0, false, B0,
                                                   (short)0, acc0, false, false);
      acc1 = __builtin_amdgcn_wmma_f32_16x16x4_f32(false, A1, false, B1,
                                                   (short)0, acc1, false, false);
    }
    v8f D = acc0 + acc1;  // D[m][n] == r[m] for every column n

    // Lag 16 by lane-parallel reduction (144 products).
    float p16 = 0.0f;
#pragma unroll
    for (int t = 0; t < 5; ++t) {
      int idx = lane + 32 * t;
      if (idx < FRAME - ORDER) p16 += fr[idx] * fr[idx + ORDER];
    }
#pragma unroll
    for (int s = 16; s >= 1; s >>= 1) p16 += __shfl_xor(p16, s, 32);

    // Broadcast r[0..15] out of D: C/D layout puts rows 0..7 in lane 0 (col 0)
    // and rows 8..15 in lane 16, element index = row & 7.
    float r[ORDER + 1];
#pragma unroll
    for (int q = 0; q < 16; ++q) r[q] = __shfl(D[q & 7], (q >> 3) * 16, 32);
    r[ORDER] = p16;

    // ---------------- Levinson-Durbin (redundant per lane, all-register) ----
    float a[ORDER + 1];
    a[0] = 1.0f;
#pragma unroll
    for (int q = 1; q <= ORDER; ++q) a[q] = 0.0f;
    float e = (r[0] != 0.0f) ? r[0] : EPS_F;
#pragma unroll
    for (int i = 1; i <= ORDER; ++i) {
      float accv = r[i];
#pragma unroll
      for (int j = 1; j < i; ++j) accv -= a[j] * r[i - j];
      float k = (e != 0.0f) ? (accv / e) : 0.0f;
      float anew[ORDER + 1];
#pragma unroll
      for (int j = 1; j < i; ++j) anew[j] = a[j] - k * a[i - j];
#pragma unroll
      for (int j = 1; j < i; ++j) a[j] = anew[j];
      a[i] = k;
      e = fmaxf(e * (1.0f - k * k), EPS_F);
    }

    // ---------------- FIR: res[n] = sum_k a[k] * f[n-k] ----------------------
    float* go = out + (long long)frame * FRAME;
#pragma unroll
    for (int t = 0; t < 5; ++t) {
      int n = lane + 32 * t;
      float s = 0.0f;
#pragma unroll
      for (int k = 0; k <= ORDER; ++k) s += a[k] * fr[n - k];
      go[n] = s;
    }
  }
}

extern "C" void kernel_launch(void* const* d_in, const int* in_sizes, int n_in,
                              void* d_out, int out_size, void* d_ws, size_t ws_size,
                              hipStream_t stream) {
  (void)n_in; (void)out_size; (void)d_ws; (void)ws_size;
  const float* x = (const float*)d_in[0];
  float* out = (float*)d_out;
  const int total = in_sizes[0];            // B * T = 20,480,000
  const int n_frames = total / FRAME;       // 128,000
  const int blocks = (n_frames + WAVES_PER_BLOCK - 1) / WAVES_PER_BLOCK;
  lpc_residual_kernel<<<blocks, WAVES_PER_BLOCK * 32, 0, stream>>>(x, out, n_frames);
}